// Loss_for_localization_83408264888736
// MI455X (gfx1250) — compile-verified
//
#include <hip/hip_runtime.h>
#include <stdint.h>

// ---------------- problem constants (from reference) ----------------
#define B_      64
#define A_      9
#define HW_     16384            // 128*128
#define C_      1000
#define NCELL   (B_ * A_ * HW_)  // 9,437,184 cells
// ---------------- tiling ----------------
#define CHUNK            128                 // cells per wave-stage (divides HW_)
#define NSTR             10                  // obj(1) + gt(5) + locs(4) rows
#define BUF_FLOATS       (NSTR * CHUNK)      // 1280 floats per buffer
#define WAVES_PER_BLOCK  4
#define BLOCK_THREADS    128
#define GRID_BLOCKS      1024
#define TOTAL_WAVES      (GRID_BLOCKS * WAVES_PER_BLOCK)   // 4096
#define NSTAGES          (NCELL / CHUNK)                   // 73728 = 4096 * 18

typedef uint32_t u32x4 __attribute__((ext_vector_type(4)));
typedef int      i32x8 __attribute__((ext_vector_type(8)));
typedef int      i32x4 __attribute__((ext_vector_type(4)));
typedef float    v2f   __attribute__((ext_vector_type(2)));
typedef float    v8f   __attribute__((ext_vector_type(8)));

// Build a CDNA5 TDM descriptor (D#) for a 2-D f32 tile and issue
// tensor_load_to_lds (6-arg clang-23 variant). Groups per CDNA5 ISA ch.8:
//  g0: [1:0]=count=1, [63:32]=lds_addr, [120:64]=global_addr, [127:126]=type=2
//  g1: [17:16]=data_size=2 (4B), tensor_dim0/1 (huge -> no OOB),
//      tile_dim0=[127:112], tile_dim1=[143:128], dim0_stride=[207:160]
__device__ __forceinline__ void tdm_load_2d(uint32_t lds_byte,
                                            const float* gptr,
                                            uint32_t tile_w_elems,
                                            uint32_t tile_h,
                                            uint32_t row_stride_elems)
{
    uint64_t ga = (uint64_t)(uintptr_t)gptr;
    const uint32_t td0 = 0x40000000u;   // tensor_dim0: huge -> never OOB
    const uint32_t td1 = 0x40000000u;   // tensor_dim1: huge -> never OOB

    u32x4 g0;
    g0[0] = 1u;                                          // count=1, user D#
    g0[1] = lds_byte;                                    // lds_addr (bytes)
    g0[2] = (uint32_t)ga;                                // global_addr[31:0]
    g0[3] = ((uint32_t)(ga >> 32) & 0x01FFFFFFu)         // global_addr[56:32]
            | (2u << 30);                                // type = 2 ("image")

    i32x8 g1;
    g1[0] = (int)(2u << 16);                             // data_size = 4 bytes
    g1[1] = (int)((td0 & 0xFFFFu) << 16);                // tensor_dim0[15:0]
    g1[2] = (int)((td0 >> 16) | ((td1 & 0xFFFFu) << 16));// td0[31:16] | td1[15:0]
    g1[3] = (int)((td1 >> 16) | (tile_w_elems << 16));   // td1[31:16] | tile_dim0
    g1[4] = (int)(tile_h & 0xFFFFu);                     // tile_dim1 (tile_dim2=0)
    g1[5] = (int)row_stride_elems;                       // tensor_dim0_stride[31:0]
    g1[6] = 0;
    g1[7] = 0;

    i32x4 gz4 = {0, 0, 0, 0};                            // groups 2/3 unused (2-D)
    i32x8 gz8 = {0, 0, 0, 0, 0, 0, 0, 0};                // extra group (clang-23 form)
    __builtin_amdgcn_tensor_load_to_lds(g0, g1, gz4, gz4, gz8, 0);
}

// Issue the 3 TDM copies for one 128-cell stage into an LDS buffer laid out as
// 10 contiguous rows of 128 floats: [obj][gt.mask][gt.c0..c3][loc.c0..c3]
__device__ __forceinline__ void issue_stage(uint32_t stg, uint32_t lds_byte,
                                            const float* __restrict__ objects,
                                            const float* __restrict__ locs,
                                            const float* __restrict__ gt)
{
    uint32_t cell0 = stg * CHUNK;
    uint32_t ba    = cell0 >> 14;          // (b*A + a), since CHUNK | HW_
    uint32_t hw0   = cell0 & (HW_ - 1u);
    const float* po = objects + cell0;
    const float* pg = gt   + (size_t)ba * (5u * HW_) + hw0;
    const float* pl = locs + (size_t)ba * (4u * HW_) + hw0;
    tdm_load_2d(lds_byte,                 po, CHUNK, 1, CHUNK);  // objects: 1x128
    tdm_load_2d(lds_byte + CHUNK * 4u,    pg, CHUNK, 5, HW_);    // gt: 5x128, stride 16384
    tdm_load_2d(lds_byte + CHUNK * 24u,   pl, CHUNK, 4, HW_);    // locs: 4x128, stride 16384
}

// Exact f32 wave32 sum via V_WMMA_F32_16X16X4_F32.
// A (16x4, documented layout): lane L<16 -> (M=L, K=0..1), lane L>=16 -> (M=L-16, K=2..3).
// Put value in first slot, 0 in second => row r K-sum = v[r] + v[r+16].
// B = all ones (layout-invariant). D: VGPR j holds row j (lanes 0-15) / j+8 (lanes 16-31),
// so per-lane sum of the 8 D regs gives rows 0-7 (lanes<16) / rows 8-15 (lanes>=16);
// one shfl_xor(16) completes the 32-lane total (broadcast to all lanes).
__device__ __forceinline__ float wave_sum_wmma(float v)
{
    v2f a; a[0] = v;    a[1] = 0.0f;
    v2f b; b[0] = 1.0f; b[1] = 1.0f;
    v8f c = {};
    v8f d = __builtin_amdgcn_wmma_f32_16x16x4_f32(false, a, false, b,
                                                  (short)0, c, false, false);
    float s = ((d[0] + d[1]) + (d[2] + d[3])) + ((d[4] + d[5]) + (d[6] + d[7]));
    s += __shfl_xor(s, 16, 32);
    return s;
}

extern "C" __global__ __launch_bounds__(BLOCK_THREADS)
void loss_main_kernel(const float* __restrict__ objects,
                      const float* __restrict__ locs,
                      const float* __restrict__ gt,
                      float* __restrict__ ws)
{
    __shared__ __attribute__((aligned(16))) float lds[WAVES_PER_BLOCK * 2 * BUF_FLOATS];
    __shared__ float red[WAVES_PER_BLOCK * 3];

    const uint32_t tid   = threadIdx.x;
    const uint32_t wid   = tid >> 5;
    const uint32_t lane  = tid & 31u;
    const uint32_t gwave = blockIdx.x * WAVES_PER_BLOCK + wid;

    float* wbuf0 = &lds[(wid * 2 + 0) * BUF_FLOATS];
    float* wbuf1 = &lds[(wid * 2 + 1) * BUF_FLOATS];
    const uint32_t lds0 = (uint32_t)(uintptr_t)wbuf0;  // LDS byte address (flat low 32)
    const uint32_t lds1 = (uint32_t)(uintptr_t)wbuf1;

    float s_noobj = 0.0f, s_obj = 0.0f, s_coor = 0.0f;

    uint32_t stg = gwave;
    if (stg < NSTAGES) issue_stage(stg, lds0, objects, locs, gt);

    uint32_t cur = 0;
    while (stg < NSTAGES) {
        uint32_t nxt = stg + TOTAL_WAVES;
        if (nxt < NSTAGES) {
            issue_stage(nxt, cur ? lds0 : lds1, objects, locs, gt);
            __builtin_amdgcn_s_wait_tensorcnt(3);   // front 3 (current stage) done
        } else {
            __builtin_amdgcn_s_wait_tensorcnt(0);   // drain
        }
        asm volatile("" ::: "memory");

        const float4* f = (const float4*)(cur ? wbuf1 : wbuf0);
        // each row of 128 floats = 32 float4; lane reads its float4 per row
        float4 p   = f[32 * 0 + lane];   // objects
        float4 m   = f[32 * 1 + lane];   // gt mask
        float4 g0v = f[32 * 2 + lane];   // gt coord 0..3
        float4 g1v = f[32 * 3 + lane];
        float4 g2v = f[32 * 4 + lane];
        float4 g3v = f[32 * 5 + lane];
        float4 l0v = f[32 * 6 + lane];   // locs coord 0..3
        float4 l1v = f[32 * 7 + lane];
        float4 l2v = f[32 * 8 + lane];
        float4 l3v = f[32 * 9 + lane];

#define DO_CELL(k)                                                        \
        {                                                                 \
            float pv = p.k, mv = m.k;                                     \
            float lp  = fmaxf(__logf(pv),        -100.0f);                \
            float l1p = fmaxf(__logf(1.0f - pv), -100.0f);                \
            s_noobj = fmaf(1.0f - mv, -l1p, s_noobj);                     \
            s_obj   = fmaf(mv, -lp, s_obj);                               \
            float d0 = l0v.k - g0v.k, d1 = l1v.k - g1v.k;                 \
            float d2 = l2v.k - g2v.k, d3 = l3v.k - g3v.k;                 \
            float sq = fmaf(d0, d0, fmaf(d1, d1, fmaf(d2, d2, d3 * d3))); \
            s_coor  = fmaf(mv, sq, s_coor);                               \
        }
        DO_CELL(x) DO_CELL(y) DO_CELL(z) DO_CELL(w)
#undef DO_CELL

        asm volatile("" ::: "memory");
        cur ^= 1u;
        stg = nxt;
    }

    // wave32 reduction via f32 WMMA (EXEC is all-ones here: uniform trip count)
    s_noobj = wave_sum_wmma(s_noobj);
    s_obj   = wave_sum_wmma(s_obj);
    s_coor  = wave_sum_wmma(s_coor);

    if (lane == 0) {
        red[wid * 3 + 0] = s_noobj;
        red[wid * 3 + 1] = s_obj;
        red[wid * 3 + 2] = s_coor;
    }
    __syncthreads();
    if (tid == 0) {
        float a = 0.f, b = 0.f, c = 0.f;
        for (int w = 0; w < WAVES_PER_BLOCK; ++w) {
            a += red[w * 3 + 0]; b += red[w * 3 + 1]; c += red[w * 3 + 2];
        }
        ws[blockIdx.x * 3 + 0] = a;
        ws[blockIdx.x * 3 + 1] = b;
        ws[blockIdx.x * 3 + 2] = c;
    }
}

extern "C" __global__ __launch_bounds__(256)
void loss_final_kernel(const float* __restrict__ scores,
                       const int* __restrict__ label,
                       const float* __restrict__ ws,
                       float* __restrict__ out)
{
    __shared__ float cls_red[8];
    __shared__ float pr[8 * 3];
    const int tid = threadIdx.x, wid = tid >> 5, lane = tid & 31;

    // image-class loss: mean_b ( logsumexp(scores[b]) - scores[b, label[b]] )
    float cls = 0.0f;
    for (int r = wid; r < B_; r += 8) {
        const float* s = scores + (size_t)r * C_;
        float mx = -3.4e38f;
        for (int c = lane; c < C_; c += 32) mx = fmaxf(mx, s[c]);
        for (int off = 16; off > 0; off >>= 1) mx = fmaxf(mx, __shfl_xor(mx, off, 32));
        float sum = 0.0f;
        for (int c = lane; c < C_; c += 32) sum += __expf(s[c] - mx);
        for (int off = 16; off > 0; off >>= 1) sum += __shfl_xor(sum, off, 32);
        float lse = mx + __logf(sum);
        if (lane == 0) cls += lse - s[label[r]];
    }

    // deterministic tree reduction of the 1024 block partials
    float a = 0.f, b = 0.f, c = 0.f;
    for (int i = tid; i < GRID_BLOCKS; i += 256) {
        a += ws[i * 3 + 0]; b += ws[i * 3 + 1]; c += ws[i * 3 + 2];
    }
    for (int off = 16; off > 0; off >>= 1) {
        a += __shfl_down(a, off, 32);
        b += __shfl_down(b, off, 32);
        c += __shfl_down(c, off, 32);
    }
    if (lane == 0) {
        cls_red[wid] = cls;
        pr[wid * 3 + 0] = a; pr[wid * 3 + 1] = b; pr[wid * 3 + 2] = c;
    }
    __syncthreads();
    if (tid == 0) {
        float tc = 0.f, ta = 0.f, tb = 0.f, tco = 0.f;
        for (int w = 0; w < 8; ++w) {
            tc  += cls_red[w];
            ta  += pr[w * 3 + 0];
            tb  += pr[w * 3 + 1];
            tco += pr[w * 3 + 2];
        }
        // IMG_CLASS_WEIGHT*cls_mean + (0.5*noobj + 1.0*obj + 5.0*coor)/B
        out[0] = tc / (float)B_ + (0.5f * ta + 1.0f * tb + 5.0f * tco) / (float)B_;
    }
}

extern "C" void kernel_launch(void* const* d_in, const int* in_sizes, int n_in,
                              void* d_out, int out_size, void* d_ws, size_t ws_size,
                              hipStream_t stream)
{
    (void)in_sizes; (void)n_in; (void)out_size; (void)ws_size;
    const float* objects = (const float*)d_in[0];
    const float* scores  = (const float*)d_in[1];
    const float* locs    = (const float*)d_in[2];
    const int*   label   = (const int*)d_in[3];
    const float* gt      = (const float*)d_in[4];
    float* ws = (float*)d_ws;   // 1024 blocks * 3 floats = 12 KB

    loss_main_kernel<<<GRID_BLOCKS, BLOCK_THREADS, 0, stream>>>(objects, locs, gt, ws);
    loss_final_kernel<<<1, 256, 0, stream>>>(scores, label, ws, (float*)d_out);
}